// EncoderLSTM_57664230916249
// MI455X (gfx1250) — compile-verified
//
#include <hip/hip_runtime.h>
#include <math.h>

// Problem constants (from reference): T=1024, B=64, D=512, H=512
#define T_STEPS 1024
#define BATCH   64
#define DIN     512
#define HID     512
#define G4      2048   // 4*H
#define KTOT    1024   // D + H (combined weight K)
#define NWG     32     // HID / 16 hidden-unit blocks
#define NTHREADS 256   // 8 waves: 0-3 produce Zx(t+? ahead), 4-7 recurrent

typedef float v2f __attribute__((ext_vector_type(2)));
typedef float v8f __attribute__((ext_vector_type(8)));

// ---------------------------------------------------------------------------
// FP32 WMMA 16x16x4 via inline asm (no clang builtin is exposed for the f32
// shape on gfx1250; VOP3P opcode 93 per the ISA). D == C tied tuple.
// ---------------------------------------------------------------------------
static __device__ __forceinline__ v8f wmma_f32_16x16x4(v2f a, v2f b, v8f c) {
  asm("v_wmma_f32_16x16x4_f32 %0, %1, %2, %0"
      : "+v"(c)
      : "v"(a), "v"(b));
  return c;
}

// ---------------------------------------------------------------------------
// Phase 0: build WcT[col][k] = (k < 512 ? Wx[k][col] : Wh[k-512][col])
// Column-major combined weights so B-fragments are contiguous b64 loads in k.
// ---------------------------------------------------------------------------
__global__ void transpose_weights(const float* __restrict__ Wx,
                                  const float* __restrict__ Wh,
                                  float* __restrict__ WcT) {
  int idx = blockIdx.x * blockDim.x + threadIdx.x;   // 0 .. 2048*1024-1
  int k = idx >> 11;                                  // 0..1023 (slow)
  int c = idx & (G4 - 1);                             // 0..2047 (fast -> coalesced reads)
  float v = (k < DIN) ? Wx[(size_t)k * G4 + c] : Wh[(size_t)(k - DIN) * G4 + c];
  WcT[(size_t)c * KTOT + k] = v;
}

// ---------------------------------------------------------------------------
// Persistent fused LSTM, producer/consumer wave pairing.
// Grid: 32 blocks x 256 threads. Block = one 16-hidden-unit slice (4 gates).
//   Waves 0-3 (X waves): row block = wave;   Zx(p) = x_p @ Wx -> LDS slot p&1
//   Waves 4-7 (H waves): row block = wave-4; step t = p-1:
//        z = Zx(t) [LDS] + h_{t-1} @ Wh + b -> gates -> c,h ; global step barrier
// Each SIMD hosts one X wave + one H wave, so load latency in one is hidden
// by WMMA issue from the other.
//
// WMMA f32 16x16x4 fragment layouts (ISA 7.12.2):
//   A 16x4 : lane l -> M = l&15, VGPR{0,1} = A[M][k0 + 2*(l>>4) + {0,1}]   (one b64)
//   B 4x16 : lane l -> N = l&15, VGPR{0,1} = B[k0 + 2*(l>>4) + {0,1}][N]   (one b64,
//            WcT column-major in k)
//   C 16x16: VGPR v, lane l -> row = v + 8*(l>>4), col = l&15
// ---------------------------------------------------------------------------
__global__ __launch_bounds__(NTHREADS, 1)
void lstm_persistent(const float* __restrict__ x,     // [T, B, D]
                     const float* __restrict__ c0,    // [B, H]
                     const float* __restrict__ h0,    // [B, H]
                     const float* __restrict__ bvec,  // [4H]
                     const float* __restrict__ WcT,   // [2048, 1024]
                     unsigned int* __restrict__ bar,  // [T] step counters (zeroed)
                     float* __restrict__ out) {       // c_f | h_f | ys
  float* c_final = out;
  float* h_final = out + BATCH * HID;
  float* ys      = out + 2 * BATCH * HID;             // [T, B, H]

  // Zx hand-off buffer in C-fragment order: [slot][rowblk][gate][lane][v]
  __shared__ __align__(32) float zx[2 * 4 * 4 * 32 * 8];   // 32 KB
#define ZX(slot, rw, g, lane) (&zx[((((slot) * 4 + (rw)) * 4 + (g)) * 32 + (lane)) * 8])

  const int tid  = threadIdx.x;
  const int lane = tid & 31;
  const int wave = tid >> 5;            // 0..7
  const bool isX = (wave < 4);
  const int rw   = isX ? wave : (wave - 4);   // batch row block 0..3
  const int m    = lane & 15;
  const int half = lane >> 4;
  const int koff = half * 2;
  const int r0   = rw * 16;
  const int arow = r0 + m;
  const int n    = blockIdx.x * 16 + m;       // hidden unit column for this lane

  // Combined-weight column bases per gate; X waves use k in [0,512),
  // H waves use k in [512,1024).
  const float* bx0 = WcT + (size_t)(0 * HID + n) * KTOT + koff;
  const float* bx1 = WcT + (size_t)(1 * HID + n) * KTOT + koff;
  const float* bx2 = WcT + (size_t)(2 * HID + n) * KTOT + koff;
  const float* bx3 = WcT + (size_t)(3 * HID + n) * KTOT + koff;
  const float* bh0 = bx0 + DIN;
  const float* bh1 = bx1 + DIN;
  const float* bh2 = bx2 + DIN;
  const float* bh3 = bx3 + DIN;

  // Recurrent state (H waves only)
  float cst[8], hl[8];
  float bias0 = 0.f, bias1 = 0.f, bias2 = 0.f, bias3 = 0.f;
#pragma unroll
  for (int v = 0; v < 8; ++v) { cst[v] = 0.f; hl[v] = 0.f; }
  if (!isX) {
    bias0 = bvec[0 * HID + n];
    bias1 = bvec[1 * HID + n];
    bias2 = bvec[2 * HID + n];
    bias3 = bvec[3 * HID + n];
#pragma unroll
    for (int v = 0; v < 8; ++v)
      cst[v] = c0[(size_t)(r0 + v + 8 * half) * HID + n];
  }

#pragma unroll 1
  for (int p = 0; p <= T_STEPS; ++p) {
    if (isX) {
      // ------------------ producer: Zx(p) = x_p @ Wx ------------------
      if (p < T_STEPS) {
        const float* aX = x + (size_t)p * BATCH * DIN + (size_t)arow * DIN + koff;
        v8f a0 = {0.f,0.f,0.f,0.f,0.f,0.f,0.f,0.f};
        v8f a1 = a0, a2 = a0, a3 = a0;
#pragma unroll 8
        for (int kk = 0; kk < DIN; kk += 4) {
          v2f a  = *(const v2f*)(aX  + kk);
          v2f b0 = *(const v2f*)(bx0 + kk);
          v2f b1 = *(const v2f*)(bx1 + kk);
          v2f b2 = *(const v2f*)(bx2 + kk);
          v2f b3 = *(const v2f*)(bx3 + kk);
          a0 = wmma_f32_16x16x4(a, b0, a0);
          a1 = wmma_f32_16x16x4(a, b1, a1);
          a2 = wmma_f32_16x16x4(a, b2, a2);
          a3 = wmma_f32_16x16x4(a, b3, a3);
        }
        // WMMA -> VALU/DS RAW hazard on D (ISA 7.12.1): explicit wait states,
        // compiler can't see through the inline asm.
        asm volatile("s_nop 8");
        *(v8f*)ZX(p & 1, rw, 0, lane) = a0;
        *(v8f*)ZX(p & 1, rw, 1, lane) = a1;
        *(v8f*)ZX(p & 1, rw, 2, lane) = a2;
        *(v8f*)ZX(p & 1, rw, 3, lane) = a3;
        // Pull next step's x rows toward L2/L0 (off critical path).
        if (p + 1 < T_STEPS) {
          const char* nx = (const char*)(x + (size_t)(p + 1) * BATCH * DIN +
                                         (size_t)arow * DIN);
          __builtin_prefetch(nx, 0, 1);
          __builtin_prefetch(nx + 1024, 0, 1);
        }
      }
    } else {
      // ------------------ consumer: step t = p-1 ------------------
      if (p >= 1) {
        const int t = p - 1;
        const float* hsrc = (t == 0) ? h0 : (ys + (size_t)(t - 1) * BATCH * HID);
        const float* aH = hsrc + (size_t)arow * HID + koff;

        // Accumulators start from Zx(t) produced last phase (same fragment map).
        v8f a0 = *(const v8f*)ZX(t & 1, rw, 0, lane);
        v8f a1 = *(const v8f*)ZX(t & 1, rw, 1, lane);
        v8f a2 = *(const v8f*)ZX(t & 1, rw, 2, lane);
        v8f a3 = *(const v8f*)ZX(t & 1, rw, 3, lane);
#pragma unroll 8
        for (int kk = 0; kk < HID; kk += 4) {
          v2f a  = *(const v2f*)(aH  + kk);
          v2f b0 = *(const v2f*)(bh0 + kk);
          v2f b1 = *(const v2f*)(bh1 + kk);
          v2f b2 = *(const v2f*)(bh2 + kk);
          v2f b3 = *(const v2f*)(bh3 + kk);
          a0 = wmma_f32_16x16x4(a, b0, a0);
          a1 = wmma_f32_16x16x4(a, b1, a1);
          a2 = wmma_f32_16x16x4(a, b2, a2);
          a3 = wmma_f32_16x16x4(a, b3, a3);
        }
        asm volatile("s_nop 8");

        // Gates: c_new = sig(f)*c + sig(i)*tanh(g); h = sig(o)*tanh(c_new)
        float* yrow = ys + (size_t)t * (BATCH * HID);
#pragma unroll
        for (int v = 0; v < 8; ++v) {
          float zi = a0[v] + bias0;
          float zf = a1[v] + bias1;
          float zg = a2[v] + bias2;
          float zo = a3[v] + bias3;
          float ig = 1.0f / (1.0f + __expf(-zi));
          float fg = 1.0f / (1.0f + __expf(-zf));
          float gg = tanhf(zg);
          float og = 1.0f / (1.0f + __expf(-zo));
          float cn = fg * cst[v] + ig * gg;
          cst[v] = cn;
          float hv = og * tanhf(cn);
          hl[v] = hv;
          yrow[(size_t)(r0 + v + 8 * half) * HID + n] = hv;
        }
        __threadfence();
      }
    }

    // Phase boundary: Zx slot hand-off + (for H) publish h(t) device-wide.
    __syncthreads();
    if (tid == 4 * 32 && p >= 1) {   // one H-wave thread runs the global barrier
      const int t = p - 1;
      __hip_atomic_fetch_add(bar + t, 1u, __ATOMIC_RELEASE, __HIP_MEMORY_SCOPE_AGENT);
      unsigned int seen;
      do {
        seen = __hip_atomic_load(bar + t, __ATOMIC_ACQUIRE, __HIP_MEMORY_SCOPE_AGENT);
        if (seen < NWG) __builtin_amdgcn_s_sleep(1);
      } while (seen < NWG);
    }
    __syncthreads();
    __threadfence();
  }

  // Final state outputs (H waves own the state)
  if (!isX) {
#pragma unroll
    for (int v = 0; v < 8; ++v) {
      size_t row = (size_t)(r0 + v + 8 * half);
      c_final[row * HID + n] = cst[v];
      h_final[row * HID + n] = hl[v];
    }
  }
#undef ZX
}

// ---------------------------------------------------------------------------
// Launch: zero barrier counters, transpose weights into ws, run persistent LSTM.
// ws layout: [0,4096) = T step counters; [4096, 4096+8MB) = WcT.
// ---------------------------------------------------------------------------
extern "C" void kernel_launch(void* const* d_in, const int* in_sizes, int n_in,
                              void* d_out, int out_size, void* d_ws, size_t ws_size,
                              hipStream_t stream) {
  const float* x  = (const float*)d_in[0];
  const float* c0 = (const float*)d_in[1];
  const float* h0 = (const float*)d_in[2];
  const float* Wx = (const float*)d_in[3];
  const float* Wh = (const float*)d_in[4];
  const float* b  = (const float*)d_in[5];

  unsigned int* bar = (unsigned int*)d_ws;
  float* WcT = (float*)((char*)d_ws + 4096);

  (void)hipMemsetAsync(d_ws, 0, 4096, stream);

  int total = G4 * KTOT;  // 2,097,152 elements
  transpose_weights<<<total / 256, 256, 0, stream>>>(Wx, Wh, WcT);

  lstm_persistent<<<NWG, NTHREADS, 0, stream>>>(x, c0, h0, b, WcT, bar,
                                                (float*)d_out);
}